// LogPoolingCovLayer_10471130268053
// MI455X (gfx1250) — compile-verified
//
#include <hip/hip_runtime.h>

// ---------------------------------------------------------------------------
// Log-polar pooling for MI455X (gfx1250).
//  x: [8,64,112,112] f32  ->  out: [8,64,224,224] f32
//  Bandwidth-bound (~128.5MB -> ~5.5us @ 23.3 TB/s), so the kernel minimizes
//  per-wave instructions:
//   - 9 input rows staged in LDS via GLOBAL_LOAD_ASYNC_TO_LDS_B128
//     (zero-padded guard columns -> no bounds checks in the inner loop)
//   - the K=81 x N=16 bin contraction runs on V_WMMA_F32_16X16X4_F32 in
//     exact f32; the entire B matrix (bin masks * 1/count) is built at
//     COMPILE TIME into a __constant__ table already in WMMA lane layout,
//     fetched as one global_load_b64 per K-chunk off a pinned base pointer
//     (asm-laundered so the compiler uses immediate offsets, not per-chunk
//     s_get_pc/s_add_nc_u64 rematerialization)
//   - result transposed through LDS, coalesced float4 stores
// ---------------------------------------------------------------------------

typedef __attribute__((ext_vector_type(2))) float v2f;
typedef __attribute__((ext_vector_type(8))) float v8f;
typedef int gv4i __attribute__((vector_size(16)));  // matches builtin's V4i

#define __AS1 __attribute__((address_space(1)))
#define __AS3 __attribute__((address_space(3)))

#if __has_builtin(__builtin_amdgcn_global_load_async_to_lds_b128) && \
    __has_builtin(__builtin_amdgcn_s_wait_asynccnt)
#define USE_ASYNC_LDS 1
#else
#define USE_ASYNC_LDS 0
#endif

#if __has_builtin(__builtin_amdgcn_wmma_f32_16x16x4_f32)
#define USE_WMMA_F32 1
#else
#define USE_WMMA_F32 0
#warning "wmma_f32_16x16x4_f32 builtin not available - scalar fallback in use"
#endif

#if USE_ASYNC_LDS
static __device__ __forceinline__ __AS1 gv4i* as_global_v4(const void* p) {
  return (__AS1 gv4i*)(unsigned long long)p;
}
static __device__ __forceinline__ __AS3 gv4i* as_lds_v4(const void* p) {
  // generic LDS flat address: low 32 bits are the LDS byte offset (ISA 10.2)
  return (__AS3 gv4i*)(unsigned)(unsigned long long)p;
}
#endif

// ---- compile-time log-polar bin map (exact reference semantics) -----------
// Reference angle logic == octant of (xp, -yp) with half-open [m*45,(m+1)*45)
// sectors (the -1e-6 epsilon pushes exact boundaries to the upper bin).
constexpr int octant(int X, int Y) {
  if (X == 0 && Y == 0) return 0;
  if (Y == 0) return (X > 0) ? 0 : 4;
  if (X == 0) return (Y > 0) ? 2 : 6;
  if (X > 0 && Y > 0) return (Y >= X) ? 1 : 0;   // 45 deg -> bin 1
  if (X < 0 && Y > 0) return (Y > -X) ? 2 : 3;   // 135 deg -> bin 3
  if (X < 0 && Y < 0) return (Y > X) ? 4 : 5;    // 225 deg -> bin 5
  return (-Y > X) ? 6 : 7;                       // 315 deg -> bin 7
}

constexpr int binOf(int kh, int kw) {            // 0-based bin index
  const int xp = kw - 4, yp = kh - 4;
  const int ac = octant(xp, -yp);
  const int pos = xp * xp + yp * yp;             // ratio=1, ini_angle=0
  const int pl = (pos <= 4) ? 0 : 1;             // dis = {4,16}, clamped
  return (ac < 4) ? ((pl == 1) ? (3 - ac) : (7 - ac))
                  : ((pl == 0) ? (ac + 4) : (ac + 8));
}

struct alignas(8) BPair { float x, y; };
struct BTab   { BPair p[21 * 32]; };
struct InvTab { float v[16]; };

constexpr BTab makeBTab() {
  BTab t{};
  int target[81] = {};
  int cnt[16] = {};
  for (int k = 0; k < 81; ++k) { target[k] = binOf(k / 9, k % 9); cnt[target[k]]++; }
  for (int kk = 0; kk < 21; ++kk)
    for (int lane = 0; lane < 32; ++lane) {
      const int n = lane & 15;
      const int k0 = 4 * kk + ((lane >> 4) << 1);   // K group per lane half
      float v0 = 0.f, v1 = 0.f;
      if (k0 < 81 && target[k0] == n && cnt[n] > 0)         v0 = 1.0f / (float)cnt[n];
      if (k0 + 1 < 81 && target[k0 + 1] == n && cnt[n] > 0) v1 = 1.0f / (float)cnt[n];
      t.p[kk * 32 + lane].x = v0;
      t.p[kk * 32 + lane].y = v1;
    }
  return t;
}

constexpr InvTab makeInvTab() {
  InvTab t{};
  int cnt[16] = {};
  for (int k = 0; k < 81; ++k) cnt[binOf(k / 9, k % 9)]++;
  for (int n = 0; n < 16; ++n) t.v[n] = (cnt[n] > 0) ? 1.0f / (float)cnt[n] : 0.f;
  return t;
}

__constant__ BTab   B_TAB   = makeBTab();
__constant__ InvTab INV_TAB = makeInvTab();

__global__ __launch_bounds__(128) void logpool_kernel(
    const float* __restrict__ x, float* __restrict__ out) {
  const int ohi = blockIdx.x;   // 0..55 output row
  const int nc  = blockIdx.y;   // 0..NC-1 (n*c folded)
  const int tid  = threadIdx.x;
  const int lane = tid & 31;
  const int wave = tid >> 5;    // 4 waves, each owns 16 owi

  __shared__ __align__(16) float s_rows[9 * 120];   // 9 rows, 4+112+4 padded
  __shared__ __align__(16) float s_out[4][16 * 20]; // per-wave transpose pad

  // ---- 1) zero guard columns, stage 9 input rows async into LDS ----------
  if (tid < 72) {                                   // 9 rows * 8 guard cols
    const int r = tid >> 3, c = tid & 7;
    s_rows[r * 120 + ((c < 4) ? c : (112 + c))] = 0.f;
  }
  const float* gbase = x + (size_t)nc * (112 * 112);
  for (int t = tid; t < 252; t += 128) {            // 9 rows * 28 float4
    const int kh  = t / 28;
    const int col = (t % 28) * 4;
    const int ih  = ohi * 2 - 4 + kh;
    float* lp = &s_rows[kh * 120 + 4 + col];        // 16B-aligned dest
    if ((unsigned)ih < 112u) {
      const float* gp = gbase + ih * 112 + col;
#if USE_ASYNC_LDS
      __builtin_amdgcn_global_load_async_to_lds_b128(as_global_v4(gp), as_lds_v4(lp), 0, 0);
#else
      *(float4*)lp = *(const float4*)gp;
#endif
    } else {
      float4 z{0.f, 0.f, 0.f, 0.f};                 // conv zero-pad rows
      *(float4*)lp = z;
    }
  }
#if USE_ASYNC_LDS
  __builtin_amdgcn_s_wait_asynccnt(0);
#endif
  __syncthreads();                                  // s_rows ready

  // ---- 2) K=81 (pad 84) x N=16 contraction on the matrix pipe ------------
  const int n     = lane & 15;                      // A: M idx / B,D: N idx
  const int delta = lane >> 4;                      // K half select (0/1)
  const int owi   = wave * 16 + n;
  const int vA    = ((owi < 56) ? owi : 55) * 2;    // clamped col base
  v8f acc = {};

#if USE_WMMA_F32
  // Pin the per-lane B-table base so each chunk is a single global_load_b64
  // with an immediate offset (no per-chunk s_get_pc/s_add rematerialization).
  typedef const __AS1 v2f* cbptr_t;
  unsigned long long bâ_bits = (unsigned long long)&B_TAB.p[lane];
  asm("" : "+v"(bâ_bits));
  cbptr_t bptr = (cbptr_t)bâ_bits;

#pragma unroll
  for (int kk = 0; kk < 21; ++kk) {
    v2f a;
#pragma unroll
    for (int i = 0; i < 2; ++i) {
      const int ka = 4 * kk + i;                    // lanes 0-15 variant
      const int kb = ka + 2;                        // lanes 16-31 variant
      const int pa = (ka < 81) ? ((ka / 9) * 120 + (ka % 9)) : 0;  // constants
      const int pb = (kb < 81) ? ((kb / 9) * 120 + (kb % 9)) : 0;
      // one v_mad + one ds_load; padded guards make this bounds-check free.
      a[i] = s_rows[pa + vA + delta * (pb - pa)];
    }
    const v2f b = bptr[kk * 32];                    // b64 load, WGP$-hot
    acc = __builtin_amdgcn_wmma_f32_16x16x4_f32(
        false, a, false, b, (short)0, acc, false, false);
  }
#else
  // scalar fallback producing the identical acc (C/D) lane layout
  (void)delta;
#pragma unroll
  for (int j = 0; j < 8; ++j) {
    const int m   = (lane < 16) ? j : (j + 8);
    const int ow2 = wave * 16 + m;
    const int vB  = ((ow2 < 56) ? ow2 : 55) * 2;
    float s = 0.f;
    for (int k = 0; k < 81; ++k) {
      const int kh = k / 9, kw = k - kh * 9;
      if (binOf(kh, kw) == n) s += s_rows[kh * 120 + vB + kw];
    }
    acc[j] = s * INV_TAB.v[n];
  }
#endif

  // ---- 3) transpose through LDS (per-wave region), coalesced stores ------
  float* so = s_out[wave];
#pragma unroll
  for (int j = 0; j < 8; ++j) {
    const int m = (lane < 16) ? j : (j + 8);        // owi within tile
    so[m * 20 + n] = acc[j];                        // bins contiguous per m
  }
  // per-wave private LDS region + in-order LDS within a wave: no barrier

  float* obase = out + (size_t)nc * (224 * 224);
#pragma unroll
  for (int t = lane; t < 64; t += 32) {             // 16 owi x 4 bin-rows
    const int m  = t >> 2;
    const int nl = t & 3;                           // NL2 block row
    const int ow = wave * 16 + m;
    if (ow < 56) {
      float4 v = *(const float4*)&so[m * 20 + nl * 4];   // bins nl*4..nl*4+3
      *(float4*)&obase[(size_t)(ohi * 4 + nl) * 224 + (size_t)ow * 4] = v;
    }
  }
}

extern "C" void kernel_launch(void* const* d_in, const int* in_sizes, int n_in,
                              void* d_out, int out_size, void* d_ws, size_t ws_size,
                              hipStream_t stream) {
  (void)n_in; (void)out_size; (void)d_ws; (void)ws_size;
  const float* x = (const float*)d_in[0];
  float* out = (float*)d_out;
  const int NC = in_sizes[0] / (112 * 112);         // 8*64 = 512
  dim3 grid(56, NC);
  logpool_kernel<<<grid, 128, 0, stream>>>(x, out);
}